// OFKD_45543833206826
// MI455X (gfx1250) — compile-verified
//
#include <hip/hip_runtime.h>
#include <stdint.h>

// Problem constants (fixed by the reference setup)
#define B_ 4
#define C_ 19
#define H_ 256
#define W_ 256
#define ROWS_ 4                       // rows pipelined per block in k_pixel

__device__ __constant__ float kInvT   = 0.25f;    // 1/TEMP
__device__ __constant__ float kEps    = 1e-8f;
__device__ __constant__ float kGamma  = 0.1f;

typedef __attribute__((ext_vector_type(4))) unsigned int v4u;
typedef __attribute__((ext_vector_type(8))) int          v8i;
typedef __attribute__((ext_vector_type(4))) int          v4i;

#if __has_builtin(__builtin_amdgcn_tensor_load_to_lds) && __has_builtin(__builtin_amdgcn_s_wait_tensorcnt)
#define TDM_OK 1
#else
#define TDM_OK 0
#endif

// ---- workspace layout (bytes) -------------------------------------------
static const size_t OFF_FG  = 0;                                   // B*H*W u8
static const size_t OFF_ENT = (size_t)B_ * H_ * W_;                // f32
static const size_t OFF_KL  = OFF_ENT + 4ull * B_ * H_ * W_;       // f32
static const size_t OFF_GG  = OFF_KL  + 4ull * B_ * H_ * W_;       // f32
static const size_t OFF_EM  = OFF_GG  + 4ull * B_ * H_ * W_;       // B int
static const size_t OFF_SUM = OFF_EM  + 4ull * B_;                 // 2 f32

// =========================================================================
__global__ void k_init(int* entmax, float* sums) {
    int t = threadIdx.x;
    if (t < B_) entmax[t] = 0;              // bits of 0.0f; entropy >= 0
    if (t == B_)     sums[0] = 0.0f;
    if (t == B_ + 1) sums[1] = 0.0f;
}

// =========================================================================
#if TDM_OK
// Issue one TDM 2D tile load: [C_ rows x W_ elems], row stride H_*W_ elems,
// 4-byte elements, into LDS at ldsDst. One tensor op -> TENSORcnt++.
__device__ __forceinline__ void tdm_row(const float* gsrc, float* ldsDst) {
    uint32_t lds32 = (uint32_t)(uintptr_t)ldsDst;  // LDS offset = low 32b of flat addr
    uint64_t ga    = (uint64_t)(uintptr_t)gsrc;
    // D# group1: data_size=4B, tensor_dim0=256, tensor_dim1=19,
    // tile_dim0=256, tile_dim1=19, tensor_dim0_stride=65536 elems.
    v8i g1 = { (int)0x00020000u, (int)0x01000000u, (int)0x00130000u,
               (int)0x01000000u, (int)19,          (int)0x00010000u, 0, 0 };
    v4i gz = {0, 0, 0, 0};
    // D# group0: count=1, lds_addr, global_addr[56:0], type=2
    v4u g0 = { 1u, lds32, (uint32_t)ga,
               (uint32_t)((ga >> 32) & 0x1FFFFFFu) | 0x80000000u };
#if defined(__clang_major__) && (__clang_major__ >= 23)
    v8i gz8 = {0, 0, 0, 0, 0, 0, 0, 0};
    __builtin_amdgcn_tensor_load_to_lds(g0, g1, gz, gz, gz8, 0);
#else
    __builtin_amdgcn_tensor_load_to_lds(g0, g1, gz, gz, 0);
#endif
}
#endif

// =========================================================================
// Per-pixel channel pass. One block = one (b, 4-row strip). Double-buffered
// TDM: row r+1 streams into the ping-pong buffer while row r is computed;
// s_wait_tensorcnt(2) throttles (TDM ops retire in-order per wave, so <=2
// outstanding means the previous row's two tensor loads are complete).
// Dynamic LDS: 2 buffers x (teacher+student) x [19x256] f32 + 256 f32 reduce.
__global__ __launch_bounds__(256) void k_pixel(const float* __restrict__ t,
                                               const float* __restrict__ s,
                                               uint8_t* __restrict__ fg,
                                               float* __restrict__ ent,
                                               float* __restrict__ kl,
                                               int* __restrict__ entmax) {
    extern __shared__ float smem[];
    const int blocksPerImg = H_ / ROWS_;               // 64
    const int b  = blockIdx.x / blocksPerImg;
    const int i0 = (blockIdx.x % blocksPerImg) * ROWS_;
    const int j  = threadIdx.x;
    float* red = smem + 2 * (2 * C_ * W_);             // after the two buffers

    float emax_local = 0.0f;

    for (int r = 0; r < ROWS_; ++r) {
        const int i = i0 + r;
        const size_t rowOff = ((size_t)b * C_ * H_ + i) * W_;
        float* ldsT = smem + (r & 1) * (2 * C_ * W_);
        float* ldsS = ldsT + C_ * W_;

#if TDM_OK
        if (threadIdx.x < 32) {        // wave 0 drives the DMA pipeline
            if (r == 0) {              // prologue: fill buffer 0
                tdm_row(t + rowOff, ldsT);
                tdm_row(s + rowOff, ldsS);
            }
            if (r + 1 < ROWS_) {       // prefetch next row into other buffer
                const size_t nOff = rowOff + W_;   // next image row (contiguous)
                float* nT = smem + ((r + 1) & 1) * (2 * C_ * W_);
                tdm_row(t + nOff, nT);
                tdm_row(s + nOff, nT + C_ * W_);
                __builtin_amdgcn_s_wait_tensorcnt(2);  // row r's 2 ops done
            } else {
                __builtin_amdgcn_s_wait_tensorcnt(0);
            }
            asm volatile("" ::: "memory");
        }
#else
        for (int c = 0; c < C_; ++c) {
            ldsT[c * W_ + j] = t[rowOff + (size_t)c * H_ * W_ + j];
            ldsS[c * W_ + j] = s[rowOff + (size_t)c * H_ * W_ + j];
        }
#endif
        __syncthreads();               // row r resident; prior compute done

        // teacher argmax (raw logits) + max for softmax
        float tmax = -3.0e38f; int amax = 0;
        for (int c = 0; c < C_; ++c) {
            float v = ldsT[c * W_ + j];
            if (v > tmax) { tmax = v; amax = c; }
        }
        float m = tmax * kInvT;
        float Z = 0.0f;
        for (int c = 0; c < C_; ++c) Z += expf(ldsT[c * W_ + j] * kInvT - m);
        float lse = m + logf(Z);

        // student log-softmax normalizer
        float smax = -3.0e38f;
        for (int c = 0; c < C_; ++c) smax = fmaxf(smax, ldsS[c * W_ + j]);
        float sm = smax * kInvT;
        float sZ = 0.0f;
        for (int c = 0; c < C_; ++c) sZ += expf(ldsS[c * W_ + j] * kInvT - sm);
        float slse = sm + logf(sZ);

        // entropy + KL in one pass
        float e = 0.0f, k = 0.0f;
        for (int c = 0; c < C_; ++c) {
            float p = expf(ldsT[c * W_ + j] * kInvT - lse);
            e -= p * logf(fmaxf(p, kEps));
            float slogp = ldsS[c * W_ + j] * kInvT - slse;
            k += p * (logf(p + kEps) - slogp);
        }

        const size_t pix = ((size_t)b * H_ + i) * W_ + j;
        fg[pix]  = (uint8_t)(amax != 0);
        ent[pix] = e;
        kl[pix]  = k;
        emax_local = fmaxf(emax_local, e);

        __syncthreads();               // all lanes done reading buf[r&1]
    }

    // per-batch entropy max: block tree-reduce then one atomic
    red[j] = emax_local;
    __syncthreads();
    for (int st = 128; st > 0; st >>= 1) {
        if (j < st) red[j] = fmaxf(red[j], red[j + st]);
        __syncthreads();
    }
    if (j == 0) atomicMax(entmax + b, __float_as_int(red[0]));
}

// =========================================================================
// Vertical distance per column (forward/backward scan), squared + clamped.
__global__ __launch_bounds__(256) void k_vert(const uint8_t* __restrict__ fg,
                                              float* __restrict__ gg) {
    const int b = blockIdx.x;
    const int j = threadIdx.x;
    const size_t base = (size_t)b * H_ * W_ + j;
    int last = -100000;
    for (int i = 0; i < H_; ++i) {
        size_t idx = base + (size_t)i * W_;
        if (i + 8 < H_) __builtin_prefetch(fg + idx + 8 * W_, 0, 0);
        if (fg[idx]) last = i;
        gg[idx] = (float)(i - last);
    }
    int nxt = 100000;
    for (int i = H_ - 1; i >= 0; --i) {
        size_t idx = base + (size_t)i * W_;
        if (fg[idx]) nxt = i;
        float g = fminf(gg[idx], (float)(nxt - i));
        g = fminf(g, 1000.0f);              // beyond RADIUS: g^2 > 100 -> masked
        gg[idx] = g * g;
    }
}

// =========================================================================
// Horizontal 21-tap window min (exact within the mask radius), weights,
// boundary via 4-neighbor erosion, confidence factor, global weighted sums.
__global__ __launch_bounds__(256) void k_row(const uint8_t* __restrict__ fg,
                                             const float* __restrict__ gg,
                                             const float* __restrict__ ent,
                                             const float* __restrict__ kl,
                                             const int* __restrict__ entmax,
                                             float* __restrict__ sums) {
    __shared__ float ggS[W_];
    __shared__ float red[W_];
    const int bi = blockIdx.x;
    const int b  = bi >> 8;
    const int i  = bi & (H_ - 1);
    const int j  = threadIdx.x;
    const size_t row = (size_t)bi * W_;

    ggS[j] = gg[row + j];
    __syncthreads();

    float d2 = 1.0e12f;
#pragma unroll
    for (int dj = -10; dj <= 10; ++dj) {
        int jj = j + dj;
        if (jj >= 0 && jj < W_)
            d2 = fminf(d2, (float)(dj * dj) + ggS[jj]);
    }
    float maskf = (d2 <= 100.0f) ? 1.0f : 0.0f;      // dist <= RADIUS
    float wdist = expf(-d2 * 0.02f) * maskf;         // exp(-d2 / (2*sigma^2)), sigma=5

    bool f  = fg[row + j] != 0;
    bool er = f;
    er = er && (i > 0      ? (fg[row + j - W_] != 0) : false);
    er = er && (i < H_ - 1 ? (fg[row + j + W_] != 0) : false);
    er = er && (j > 0      ? (fg[row + j - 1]  != 0) : false);
    er = er && (j < W_ - 1 ? (fg[row + j + 1]  != 0) : false);
    float boundary = (f && !er) ? 1.0f : 0.0f;

    float weight = wdist * (1.0f + boundary) * maskf;   // lambda_b = 1
    float em   = __int_as_float(entmax[b]);
    float conf = kGamma + (1.0f - kGamma) * (1.0f - ent[row + j] / (em + kEps));
    float tw   = weight * conf;
    float twkl = tw * kl[row + j];

    // block reduce sum(tw) then sum(tw*kl)
    red[j] = tw;
    __syncthreads();
    for (int st = 128; st > 0; st >>= 1) {
        if (j < st) red[j] += red[j + st];
        __syncthreads();
    }
    float s0 = red[0];
    __syncthreads();
    red[j] = twkl;
    __syncthreads();
    for (int st = 128; st > 0; st >>= 1) {
        if (j < st) red[j] += red[j + st];
        __syncthreads();
    }
    if (j == 0) {
        atomicAdd(sums + 0, s0);
        atomicAdd(sums + 1, red[0]);
    }
}

// =========================================================================
__global__ void k_final(const float* __restrict__ sums, float* __restrict__ out) {
    if (threadIdx.x == 0 && blockIdx.x == 0)
        out[0] = 16.0f * sums[1] / (sums[0] + kEps);   // TEMP^2 * loss
}

// =========================================================================
extern "C" void kernel_launch(void* const* d_in, const int* in_sizes, int n_in,
                              void* d_out, int out_size, void* d_ws, size_t ws_size,
                              hipStream_t stream) {
    const float* stud  = (const float*)d_in[0];   // student_logits
    const float* teach = (const float*)d_in[1];   // teacher_logits
    char* ws = (char*)d_ws;
    uint8_t* fg   = (uint8_t*)(ws + OFF_FG);
    float*   ent  = (float*)(ws + OFF_ENT);
    float*   kl   = (float*)(ws + OFF_KL);
    float*   gg   = (float*)(ws + OFF_GG);
    int*     em   = (int*)(ws + OFF_EM);
    float*   sums = (float*)(ws + OFF_SUM);

    // dynamic LDS: 2 ping-pong buffers x (T+S) x [19x256] f32 + 256 f32
    const size_t ldsBytes = (2ull * (2 * C_ * W_) + W_) * sizeof(float);  // 78848

    k_init <<<1, 64, 0, stream>>>(em, sums);
    k_pixel<<<B_ * (H_ / ROWS_), 256, ldsBytes, stream>>>(teach, stud, fg, ent, kl, em);
    k_vert <<<B_, 256, 0, stream>>>(fg, gg);
    k_row  <<<B_ * H_, 256, 0, stream>>>(fg, gg, ent, kl, em, sums);
    k_final<<<1, 1, 0, stream>>>(sums, (float*)d_out);
}